// LIF_ASC_34531537060428
// MI455X (gfx1250) — compile-verified
//
#include <hip/hip_runtime.h>

typedef __attribute__((ext_vector_type(2))) float v2f;
typedef __attribute__((ext_vector_type(8))) float v8f;

#define T_STEPS 4096
#define BATCH   256
#define NN      12
#define THR     30.0f
#define GAIN    1.75f
#define RCONST  1.1f

// One wave32 per 16-batch tile. State (v, s, I_add) lives in the WMMA f32
// 16x16 C/D layout: reg r, lane l -> element (m = 8*(l/16) + r, n = l%16),
// m = batch row within tile, n = neuron (12 real + 4 zero padding).
// K dimension of the recurrent matmul spans only the 12 real neurons ->
// three V_WMMA_F32_16X16X4_F32 slabs (K=0..11); the K=12..15 slab is
// identically zero and is omitted.
__global__ __launch_bounds__(32) void lif_asc_wmma_kernel(
    const float* __restrict__ x,   // (T, B, 12)
    const float* __restrict__ w,   // (12, 12)
    const float* __restrict__ Gp,  // (12,)
    const float* __restrict__ ELp, // (12,)
    const float* __restrict__ tmp, // (12,) tau_m
    const float* __restrict__ tsp, // (12,) tau_s
    const float* __restrict__ fIp, // (12,)
    const int*   __restrict__ ntp, // (12,) neuron_types
    float* __restrict__ out)       // (T, B, 12)
{
    __shared__ float ldsA[16 * 17];   // A-transpose staging (padded stride)
    __shared__ float ldsO[16 * 12 + 16 * 16];  // flat out tile + pad dump zone

    const int lane = threadIdx.x & 31;
    const int h    = lane >> 4;       // lane half
    const int n    = lane & 15;       // neuron column handled by this lane
    const bool act = (n < NN);
    const int b0   = blockIdx.x * 16;

    // Per-neuron constants (pad lanes get values that keep state identically 0)
    const float Gc  = act ? Gp[n]  : 0.0f;
    const float EL  = act ? ELp[n] : 0.0f;
    const float tm  = act ? tmp[n] : 1.0f;
    const float ts  = act ? tsp[n] : 1.0f;
    const float fI  = act ? fIp[n] : 0.0f;

    // Folded update constants; laundered so the divide expansions can never
    // be rematerialized inside the loop.
    float A1        = Gc / tm;                 // G / tau_m
    float A2        = (THR - EL) * RCONST / tm;// normR / tau_m
    float inv_ts    = 1.0f / ts;
    float inv_thrEL = 1.0f / (THR - EL);
    asm volatile("" : "+v"(A1), "+v"(A2), "+v"(inv_ts), "+v"(inv_thrEL));
    const float gmask = act ? GAIN : 0.0f;     // zeroes pad x columns, branchless

    // Branchless staging bases (per-lane constants, single address register each)
    const int ncl   = act ? n : 0;                   // clamped x column
    const int obase = act ? n : (16 * NN + n);       // flat out col or dump zone
    const int aBase = (8 * h) * 17 + n;              // ldsA store base
    const int gBase = (lane & 15) * 17 + 2 * h;      // ldsA gather base
    const int oStB  = (8 * h) * NN + obase;          // ldsO store base
    const int xBase = (8 * h) * NN + ncl;            // x load base (elements)

    // W_eff = w * nt[:,None] * (1 - eye), pre-packed into the B-matrix layout
    // for V_WMMA_F32_16X16X4_F32: chunk c, reg j, half h -> (K = 4c+2h+j, N = n)
    // Only chunks 0..2 (K = 0..11) are non-zero.
    v2f wb[3];
#pragma unroll
    for (int c = 0; c < 3; ++c) {
#pragma unroll
        for (int j = 0; j < 2; ++j) {
            const int K = 4 * c + 2 * h + j;
            float val = 0.0f;
            if (K < NN && act && K != n)
                val = w[K * NN + n] * (float)ntp[K];
            wb[c][j] = val;
        }
    }

    // Recurrent state in C-layout registers
    float v[8], s[8], Ia[8];
#pragma unroll
    for (int r = 0; r < 8; ++r) { v[r] = 0.0f; s[r] = 0.0f; Ia[r] = 0.0f; }

    // Preload x(t=0), fully unpredicated (base + immediate offsets)
    float xc[8];
    {
        const float* x0 = x + (size_t)b0 * NN + xBase;
#pragma unroll
        for (int r = 0; r < 8; ++r) xc[r] = x0[r * NN];
    }

    for (int t = 0; t < T_STEPS; ++t) {
        // ---- software-pipelined load of x(t+1) (off the critical path) ----
        const int tn = (t + 1 < T_STEPS) ? (t + 1) : t;
        const float* xnp = x + ((size_t)tn * BATCH + b0) * NN + xBase;
        if (t + 8 < T_STEPS)
            __builtin_prefetch(xnp + (size_t)7 * BATCH * NN + lane * 24, 0, 1);
        float xn[8];
#pragma unroll
        for (int r = 0; r < 8; ++r) xn[r] = xnp[r * NN];

        // ---- stage A source and previous step's output in one barrier window ----
        // out(t-1) value is just s*tau_s (s still holds step t-1's s_new).
#pragma unroll
        for (int r = 0; r < 8; ++r)
            ldsA[aBase + r * 17] = 0.5f * (Ia[r] + s[r]);
#pragma unroll
        for (int r = 0; r < 8; ++r)
            ldsO[oStB + r * NN] = s[r] * ts;
        __syncthreads();

        // Gather A chunks in A-matrix layout:
        // chunk c, reg j, lane -> (M = lane%16, K = 4c + 2h + j); K<12 only.
        v2f a[3];
#pragma unroll
        for (int c = 0; c < 3; ++c) {
#pragma unroll
            for (int j = 0; j < 2; ++j)
                a[c][j] = ldsA[gBase + 4 * c + j];
        }

        // Coalesced store of out(t-1): contiguous gather (one base, imm offsets)
        // then 768B/wave as 3x global_store_b64 per lane.
        if (t > 0) {
            const v2f* og = reinterpret_cast<const v2f*>(&ldsO[lane * 6]);
            float* op = out + ((size_t)(t - 1) * BATCH + b0) * NN + (size_t)lane * 6;
#pragma unroll
            for (int i = 0; i < 3; ++i)
                *reinterpret_cast<v2f*>(op + 2 * i) = og[i];
        }

        // ---- C init = INPUT_GAIN * x_t (branchless; pad columns zeroed) ----
        v8f acc;
#pragma unroll
        for (int r = 0; r < 8; ++r) acc[r] = gmask * xc[r];

        // I = A @ W_eff + C   (K = 12 in three K=4 slabs)
        acc = __builtin_amdgcn_wmma_f32_16x16x4_f32(false, a[0], false, wb[0], (short)0, acc, false, false);
        acc = __builtin_amdgcn_wmma_f32_16x16x4_f32(false, a[1], false, wb[1], (short)0, acc, false, false);
        acc = __builtin_amdgcn_wmma_f32_16x16x4_f32(false, a[2], false, wb[2], (short)0, acc, false, false);

        // ---- LIF state update (pure per-lane VALU) ----
#pragma unroll
        for (int r = 0; r < 8; ++r) {
            const float I   = acc[r];
            const float dv  = A1 * (EL - v[r]) + A2 * I;
            const float vn  = v[r] + dv;
            const float gat = fminf(fmaxf(vn * (1.0f / THR), 0.0f), 1.0f);
            const float dcl = fminf(fmaxf(dv * inv_thrEL, 0.0f), 1.0f);
            const float sn  = fmaf(inv_ts, gat * dcl - s[r], s[r]);
            const bool  sp  = (vn >= THR);
            v[r]  = sp ? EL : vn;
            s[r]  = sn;
            Ia[r] = fmaf(fI, (sp ? 1.0f : 0.0f) - Ia[r], Ia[r]);
            xc[r] = xn[r];
        }
    }

    // ---- flush out(T-1) ----
    __syncthreads();
#pragma unroll
    for (int r = 0; r < 8; ++r)
        ldsO[oStB + r * NN] = s[r] * ts;
    __syncthreads();
    {
        const v2f* og = reinterpret_cast<const v2f*>(&ldsO[lane * 6]);
        float* op = out + ((size_t)(T_STEPS - 1) * BATCH + b0) * NN + (size_t)lane * 6;
#pragma unroll
        for (int i = 0; i < 3; ++i)
            *reinterpret_cast<v2f*>(op + 2 * i) = og[i];
    }
}

extern "C" void kernel_launch(void* const* d_in, const int* in_sizes, int n_in,
                              void* d_out, int out_size, void* d_ws, size_t ws_size,
                              hipStream_t stream) {
    (void)in_sizes; (void)n_in; (void)d_ws; (void)ws_size; (void)out_size;
    const float* x   = (const float*)d_in[0];
    const float* w   = (const float*)d_in[1];
    const float* G   = (const float*)d_in[2];
    const float* EL  = (const float*)d_in[3];
    const float* tm  = (const float*)d_in[4];
    const float* tss = (const float*)d_in[5];
    const float* fI  = (const float*)d_in[6];
    const int*   nt  = (const int*)d_in[7];
    float* out = (float*)d_out;

    dim3 grid(BATCH / 16);   // 16 waves, one per 16-batch tile
    dim3 block(32);          // one wave32
    lif_asc_wmma_kernel<<<grid, block, 0, stream>>>(x, w, G, EL, tm, tss, fI, nt, out);
}